// GlobalModel_TWO_10393820857014
// MI455X (gfx1250) — compile-verified
//
#include <hip/hip_runtime.h>

// Problem constants (match reference)
#define NUM_GRAPHS 512
#define N_NODES    100000
#define N_EDGES    1600000
#define D_NODE     128
#define D_EDGE     32
#define D_IN       160   // D_NODE + D_EDGE
#define D_OUT      64

typedef __attribute__((ext_vector_type(2))) float v2f;
typedef __attribute__((ext_vector_type(8))) float v8f;

// ---------------------------------------------------------------------------
// Kernel 0: zero the [512 x 160] accumulator in workspace
// ---------------------------------------------------------------------------
__global__ void zero_acc_kernel(float* __restrict__ p, int n) {
    int i = blockIdx.x * blockDim.x + threadIdx.x;
    if (i < n) p[i] = 0.0f;
}

// ---------------------------------------------------------------------------
// Kernel 1: node segment-sum. batch is SORTED, so each block walks a
// contiguous row range, accumulates per-column in a register while the
// segment id is constant, and only touches global atomics at boundaries.
// blockDim = 128 (one thread per node feature column).
// ---------------------------------------------------------------------------
__global__ void node_agg_kernel(const float* __restrict__ x,
                                const long long* __restrict__ batch,
                                float* __restrict__ agg,
                                int rows_per_block) {
    const int t  = threadIdx.x;                       // column 0..127
    int r0 = blockIdx.x * rows_per_block;
    int r1 = r0 + rows_per_block;
    if (r1 > N_NODES) r1 = N_NODES;
    if (r0 >= r1) return;

    int   cur = (int)batch[r0];
    float acc = 0.0f;
    for (int r = r0; r < r1; ++r) {
        int g = (int)batch[r];                        // uniform across wave
        if (g != cur) {
            atomicAdd(&agg[(size_t)cur * D_IN + t], acc);
            acc = 0.0f;
            cur = g;
        }
        acc += x[(size_t)r * D_NODE + t];             // coalesced stream
    }
    atomicAdd(&agg[(size_t)cur * D_IN + t], acc);
}

// ---------------------------------------------------------------------------
// Kernel 2: edge segment-sum via per-block LDS accumulator [512 x 32] (64 KB).
// Random scatter goes to LDS atomics (ds_add_f32); one global atomic flush
// per block at the end. blockDim = 256: 8 edges in flight, 32 cols each.
// ---------------------------------------------------------------------------
__global__ void edge_agg_kernel(const float* __restrict__ edge_attr,
                                const long long* __restrict__ col,
                                const long long* __restrict__ batch,
                                float* __restrict__ agg,
                                int edges_per_block) {
    __shared__ float lacc[NUM_GRAPHS * D_EDGE];       // 64 KB of the 320 KB WGP LDS

    const int t = threadIdx.x;                        // 0..255
    for (int i = t; i < NUM_GRAPHS * D_EDGE; i += 256) lacc[i] = 0.0f;
    __syncthreads();

    const int c   = t & 31;                           // edge feature column
    const int sub = t >> 5;                           // which of 8 concurrent edges
    int e0 = blockIdx.x * edges_per_block;
    int e1 = e0 + edges_per_block;
    if (e1 > N_EDGES) e1 = N_EDGES;

    for (int e = e0 + sub; e < e1; e += 8) {
        int g   = (int)batch[(int)col[e]];            // uniform per 32-lane group
        float v = edge_attr[(size_t)e * D_EDGE + c];  // coalesced stream
        atomicAdd(&lacc[g * D_EDGE + c], v);          // LDS float atomic
    }
    __syncthreads();

    // Flush block-private accumulator into global agg columns [128..160)
    for (int i = t; i < NUM_GRAPHS * D_EDGE; i += 256) {
        float v = lacc[i];
        if (v != 0.0f)
            atomicAdd(&agg[(size_t)(i >> 5) * D_IN + D_NODE + (i & 31)], v);
    }
}

// ---------------------------------------------------------------------------
// Kernel 3: out[512,64] = agg[512,160] @ W[160,64] + b  via V_WMMA_F32_16X16X4_F32.
// One wave per 16x16 output tile; grid = (512/16, 64/16) = (32, 4).
// Lane layouts per CDNA5 ISA 7.12.2:
//   A 16x4 f32 : lanes 0-15 -> K={0,1} in vgpr{0,1}; lanes 16-31 -> K={2,3}
//   B 4x16 f32 : vgpr j, lanes 0-15 -> K=j; lanes 16-31 -> K=j+2 ; N = lane&15
//   C/D 16x16  : vgpr v, lanes 0-15 -> M=v; lanes 16-31 -> M=v+8 ; N = lane&15
// blockDim = 32 so EXEC is all-1s (WMMA requirement).
// ---------------------------------------------------------------------------
__global__ void gemm_wmma_kernel(const float* __restrict__ agg,
                                 const float* __restrict__ W,
                                 const float* __restrict__ bias,
                                 float* __restrict__ out) {
    const int lane = threadIdx.x;       // 0..31
    const int half = lane >> 4;         // 0 or 1
    const int ln   = lane & 15;
    const int mt   = blockIdx.x;        // 0..31
    const int nt   = blockIdx.y;        // 0..3
    const int m    = mt * 16 + ln;      // A-matrix row for this lane
    const int n    = nt * 16 + ln;      // B/D column for this lane

    v8f c = {};
    #pragma unroll 4
    for (int k0 = 0; k0 < D_IN; k0 += 4) {
        const int ka = k0 + half * 2;
        v2f a, b;
        a.x = agg[(size_t)m * D_IN + ka + 0];
        a.y = agg[(size_t)m * D_IN + ka + 1];
        b.x = W[(size_t)(ka + 0) * D_OUT + n];
        b.y = W[(size_t)(ka + 1) * D_OUT + n];
        // (neg_a, A, neg_b, B, c_mod, C, reuse_a, reuse_b)
        c = __builtin_amdgcn_wmma_f32_16x16x4_f32(
                false, a, false, b, (short)0, c, false, false);
    }

    const float bv = bias[n];
    #pragma unroll
    for (int v = 0; v < 8; ++v) {
        const int row = mt * 16 + half * 8 + v;
        out[(size_t)row * D_OUT + n] = c[v] + bv;
    }
}

// ---------------------------------------------------------------------------
// Launcher
// ---------------------------------------------------------------------------
extern "C" void kernel_launch(void* const* d_in, const int* in_sizes, int n_in,
                              void* d_out, int out_size, void* d_ws, size_t ws_size,
                              hipStream_t stream) {
    const float*     x          = (const float*)d_in[0];
    const long long* edge_index = (const long long*)d_in[1]; // [2, N_EDGES]
    const float*     edge_attr  = (const float*)d_in[2];
    /* d_in[3] = u : unused by the reference */
    const long long* batch      = (const long long*)d_in[4];
    const float*     W          = (const float*)d_in[5];
    const float*     bias       = (const float*)d_in[6];
    float*           out        = (float*)d_out;

    float* agg = (float*)d_ws;                 // [512 x 160] = 320 KiB scratch
    const long long* col = edge_index + N_EDGES;  // second row of edge_index

    // 0) zero accumulator (d_ws is poisoned; must re-init every call)
    {
        const int n = NUM_GRAPHS * D_IN;
        zero_acc_kernel<<<(n + 255) / 256, 256, 0, stream>>>(agg, n);
    }

    // 1) node aggregate: 512 blocks x 128 threads, contiguous sorted segments
    {
        const int nblocks = 512;
        const int rpb = (N_NODES + nblocks - 1) / nblocks;
        node_agg_kernel<<<nblocks, 128, 0, stream>>>(x, batch, agg, rpb);
    }

    // 2) edge aggregate: 256 blocks x 256 threads, 64KB LDS accumulator each
    {
        const int eblocks = 256;
        const int epb = (N_EDGES + eblocks - 1) / eblocks;
        edge_agg_kernel<<<eblocks, 256, 0, stream>>>(edge_attr, col, batch, agg, epb);
    }

    // 3) tiny GEMM + bias with f32 WMMA: one wave per 16x16 tile
    gemm_wmma_kernel<<<dim3(32, 4), 32, 0, stream>>>(agg, W, bias, out);
}